// GNN_14817637171441
// MI455X (gfx1250) — compile-verified
//
#include <hip/hip_runtime.h>
#include <hip/hip_bf16.h>
#include <math.h>

#define FEAT   128
#define ALPHA  0.2f
#define BN_EPS 1e-5f

typedef float v2f __attribute__((ext_vector_type(2)));
typedef float v8f __attribute__((ext_vector_type(8)));

// ---------------- helpers ----------------
__device__ __forceinline__ void atomAddF(float* p, float v) {
    __hip_atomic_fetch_add(p, v, __ATOMIC_RELAXED, __HIP_MEMORY_SCOPE_AGENT);
}
__device__ __forceinline__ void atomMaxU(unsigned* p, unsigned v) {
    __hip_atomic_fetch_max(p, v, __ATOMIC_RELAXED, __HIP_MEMORY_SCOPE_AGENT);
}
// order-preserving float<->uint key (monotone: bigger float -> bigger key)
__device__ __forceinline__ unsigned f2key(float f) {
    unsigned u = __float_as_uint(f);
    return (u & 0x80000000u) ? ~u : (u | 0x80000000u);
}
__device__ __forceinline__ float key2f(unsigned k) {
    unsigned u = (k & 0x80000000u) ? (k & 0x7FFFFFFFu) : ~k;
    return __uint_as_float(u);
}

// ---------------- generic fill ----------------
__global__ void k_fill(float* p, float v, long long n) {
    long long i = (long long)blockIdx.x * blockDim.x + threadIdx.x;
    long long stride = (long long)gridDim.x * blockDim.x;
    for (; i < n; i += stride) p[i] = v;
}

// ---------------- WMMA GEMM: C[nrows,128] = A[nrows,128] @ B[128,128] ----------------
// Optional fused per-feature affine+ReLU on A (BN folded into layer-1 input).
// Block: 256 threads = 8 waves; block computes 16 rows x 128 cols; wave w does col tile w.
// C is bounced through LDS so global stores are coalesced b128 instead of 8 strided b32.
__global__ void __launch_bounds__(256)
k_gemm(const float* __restrict__ A, const float* __restrict__ B,
       float* __restrict__ C,
       const float* __restrict__ scale, const float* __restrict__ shift,
       int nrows) {
    __shared__ __align__(16) float As[16 * 132];   // padded stride: conflict-free frag reads
    __shared__ __align__(16) float Cs[16 * 132];   // padded stride: conflict-free frag writes
    const int tid  = threadIdx.x;
    const int wave = tid >> 5;
    const int lane = tid & 31;
    const int mbase = blockIdx.x * 16;

    // stage A tile 16x128 (coalesced float4 x2 per thread), fused BN+ReLU if requested
    {
        const int r  = tid >> 4;          // 0..15
        const int c0 = (tid & 15) * 8;    // 0..120
        int rr = mbase + r; if (rr >= nrows) rr = nrows - 1;  // clamp: keep EXEC full
        const float4* src = reinterpret_cast<const float4*>(A + (size_t)rr * FEAT + c0);
        float4 q0 = src[0], q1 = src[1];
        float v[8] = {q0.x, q0.y, q0.z, q0.w, q1.x, q1.y, q1.z, q1.w};
        if (scale) {
#pragma unroll
            for (int j = 0; j < 8; ++j) {
                float t = v[j] * scale[c0 + j] + shift[c0 + j];
                v[j] = t > 0.f ? t : 0.f;   // BN + ReLU
            }
        }
        float* dst = As + r * 132 + c0;
#pragma unroll
        for (int j = 0; j < 8; ++j) dst[j] = v[j];
    }
    __syncthreads();

    const int half  = lane >> 4;     // 0: lanes 0-15, 1: lanes 16-31
    const int l16   = lane & 15;
    const int nbase = wave * 16;

    v8f acc = {};
#pragma unroll 4
    for (int k0 = 0; k0 < FEAT; k0 += 4) {
        // A fragment 16x4: lane(l16,half) holds A[l16][k0+2*half .. +1]
        const float* ap = As + l16 * 132 + k0 + 2 * half;
        v2f a; a.x = ap[0]; a.y = ap[1];
        // B fragment 4x16: lane(l16,half) holds B[k0+2*half][nbase+l16], B[k0+2*half+1][...]
        const float* bp = B + (size_t)(k0 + 2 * half) * FEAT + nbase + l16;
        v2f b; b.x = bp[0]; b.y = bp[FEAT];
        acc = __builtin_amdgcn_wmma_f32_16x16x4_f32(false, a, false, b,
                                                    (short)0, acc, false, false);
    }

    // C fragment -> LDS: VGPR r maps to tile row r + 8*half, col nbase + l16
    {
        float* cp = Cs + (size_t)(8 * half) * 132 + nbase + l16;
#pragma unroll
        for (int r = 0; r < 8; ++r) cp[(size_t)r * 132] = acc[r];
    }
    __syncthreads();

    // coalesced b128 store of the 16x128 tile (uniform row guard only)
    {
        const int r  = tid >> 4;
        const int c0 = (tid & 15) * 8;
        const int row = mbase + r;
        if (row < nrows) {
            const float4* sp = reinterpret_cast<const float4*>(Cs + r * 132 + c0);
            float4* dp = reinterpret_cast<float4*>(C + (size_t)row * FEAT + c0);
            dp[0] = sp[0];
            dp[1] = sp[1];
        }
    }
}

// ---------------- per-node attention scores: s1 = Wh.a[:128], s2 = Wh.a[128:] ----------------
__global__ void __launch_bounds__(256)
k_scores(const float* __restrict__ Wh, const float* __restrict__ a,
         float* __restrict__ s1, float* __restrict__ s2, int n) {
    const int wave = threadIdx.x >> 5, lane = threadIdx.x & 31;
    const int node = blockIdx.x * 8 + wave;
    if (node >= n) return;
    float4 v  = reinterpret_cast<const float4*>(Wh + (size_t)node * FEAT)[lane];
    float4 aa = reinterpret_cast<const float4*>(a)[lane];
    float4 ab = reinterpret_cast<const float4*>(a + FEAT)[lane];
    float d1 = v.x*aa.x + v.y*aa.y + v.z*aa.z + v.w*aa.w;
    float d2 = v.x*ab.x + v.y*ab.y + v.z*ab.z + v.w*ab.w;
#pragma unroll
    for (int off = 16; off > 0; off >>= 1) {
        d1 += __shfl_down(d1, off, 32);
        d2 += __shfl_down(d2, off, 32);
    }
    if (lane == 0) { s1[node] = d1; s2[node] = d2; }
}

// ---------------- edge pass 1: leaky-relu score + segment max ----------------
__global__ void k_edge_max(const float* __restrict__ s1, const float* __restrict__ s2,
                           const int* __restrict__ erow, const int* __restrict__ ecol,
                           int E, int Etot,
                           float* __restrict__ eout, unsigned* __restrict__ rowmax) {
    int e = blockIdx.x * blockDim.x + threadIdx.x;
    if (e >= Etot) return;
    int r, c;
    if (e < E) { r = erow[e]; c = ecol[e]; } else { r = c = e - E; }  // self-loops
    float sc = s1[r] + s2[c];
    sc = sc > 0.f ? sc : ALPHA * sc;
    eout[e] = sc;
    atomMaxU(rowmax + r, f2key(sc));
}

// ---------------- edge pass 2: exp(e - rowmax) + segment sum ----------------
__global__ void k_edge_exp(float* __restrict__ eout, const int* __restrict__ erow,
                           int E, int Etot,
                           const unsigned* __restrict__ rowmax, float* __restrict__ rowsum) {
    int e = blockIdx.x * blockDim.x + threadIdx.x;
    if (e >= Etot) return;
    int r = (e < E) ? erow[e] : (e - E);
    float ev = expf(eout[e] - key2f(rowmax[r]));
    eout[e] = ev;
    atomAddF(rowsum + r, ev);
}

// ---------------- edge pass 3: SpMM h'[row] += (ev/rowsum[row]) * Wh[col] ----------------
// One wave per edge; L2-resident gather + fp32 atomic scatter.
__global__ void __launch_bounds__(256)
k_spmm(const float* __restrict__ eout, const float* __restrict__ rowsum,
       const int* __restrict__ erow, const int* __restrict__ ecol,
       int E, int Etot,
       const float* __restrict__ Wh, float* __restrict__ hacc) {
    const int wave = threadIdx.x >> 5, lane = threadIdx.x & 31;
    const int e = blockIdx.x * 8 + wave;
    if (e >= Etot) return;
    int r, c;
    if (e < E) { r = erow[e]; c = ecol[e]; } else { r = c = e - E; }
    const float w = eout[e] / rowsum[r];
    float4 v = reinterpret_cast<const float4*>(Wh + (size_t)c * FEAT)[lane];
    float* dst = hacc + (size_t)r * FEAT + lane * 4;
    atomAddF(dst + 0, w * v.x);
    atomAddF(dst + 1, w * v.y);
    atomAddF(dst + 2, w * v.z);
    atomAddF(dst + 3, w * v.w);
}

// ---------------- ELU in place (+ optional per-feature BN stats) ----------------
__global__ void __launch_bounds__(128)
k_elu_stats(float* __restrict__ h, int n,
            float* __restrict__ bnsum, float* __restrict__ bnsumsq, int doStats) {
    const int t  = threadIdx.x;          // feature
    const int n0 = blockIdx.x * 128;     // node chunk
    float s = 0.f, ss = 0.f;
    for (int i = 0; i < 128; ++i) {
        int node = n0 + i;
        if (node >= n) break;
        float* p = h + (size_t)node * FEAT + t;
        float v = *p;
        v = v > 0.f ? v : expm1f(v);     // ELU
        *p = v;
        s += v; ss += v * v;
    }
    if (doStats) { atomAddF(bnsum + t, s); atomAddF(bnsumsq + t, ss); }
}

// ---------------- finalize BN affine: scale = g*rsqrt(var+eps), shift = b - mean*scale ----
__global__ void k_bn_final(const float* bnsum, const float* bnsumsq,
                           const float* gamma, const float* beta,
                           float* scale, float* shift, int n) {
    int t = threadIdx.x;
    float mean = bnsum[t] / (float)n;
    float var  = bnsumsq[t] / (float)n - mean * mean;
    float sc   = gamma[t] * rsqrtf(var + BN_EPS);
    scale[t] = sc;
    shift[t] = beta[t] - mean * sc;
}

// ---------------- host ----------------
extern "C" void kernel_launch(void* const* d_in, const int* in_sizes, int n_in,
                              void* d_out, int out_size, void* d_ws, size_t ws_size,
                              hipStream_t stream) {
    const float* x     = (const float*)d_in[0];
    const float* W0    = (const float*)d_in[1];
    const float* a0    = (const float*)d_in[2];
    const float* W1    = (const float*)d_in[3];
    const float* a1    = (const float*)d_in[4];
    const float* gamma = (const float*)d_in[5];
    const float* beta  = (const float*)d_in[6];
    const int*   edges = (const int*)d_in[7];

    const int N    = in_sizes[0] / FEAT;
    const int E    = in_sizes[7] / 2;
    const int Etot = E + N;
    const int* erow = edges;
    const int* ecol = edges + E;

    char* w = (char*)d_ws;
    auto alloc = [&](size_t bytes) -> char* {
        char* p = w; w += (bytes + 255) & ~(size_t)255; return p;
    };
    float*    Wh    = (float*)alloc((size_t)N * FEAT * 4);
    float*    hacc  = (float*)alloc((size_t)N * FEAT * 4);
    float*    s1    = (float*)alloc((size_t)N * 4);
    float*    s2    = (float*)alloc((size_t)N * 4);
    unsigned* rmx   = (unsigned*)alloc((size_t)N * 4);
    float*    rsum  = (float*)alloc((size_t)N * 4);
    float*    eexp  = (float*)alloc((size_t)Etot * 4);
    float*    bnsum = (float*)alloc(FEAT * 4);
    float*    bnsq  = (float*)alloc(FEAT * 4);
    float*    bnsc  = (float*)alloc(FEAT * 4);
    float*    bnsh  = (float*)alloc(FEAT * 4);

    const int mblocks = (N + 15) / 16;
    const int nodeW   = (N + 7) / 8;
    const int edgeB   = (Etot + 255) / 256;
    const int edgeW   = (Etot + 7) / 8;
    const int eluB    = (N + 127) / 128;
    float* out = (float*)d_out;

    // ===== layer 0 =====
    k_fill<<<1024, 256, 0, stream>>>(hacc, 0.f, (long long)N * FEAT);
    k_fill<<<256, 256, 0, stream>>>((float*)rmx, 0.f, N);   // bits 0 == minimal key
    k_fill<<<256, 256, 0, stream>>>(rsum, 0.f, N);
    k_fill<<<1, 128, 0, stream>>>(bnsum, 0.f, FEAT);
    k_fill<<<1, 128, 0, stream>>>(bnsq, 0.f, FEAT);

    k_gemm<<<mblocks, 256, 0, stream>>>(x, W0, Wh, nullptr, nullptr, N);
    k_scores<<<nodeW, 256, 0, stream>>>(Wh, a0, s1, s2, N);
    k_edge_max<<<edgeB, 256, 0, stream>>>(s1, s2, erow, ecol, E, Etot, eexp, rmx);
    k_edge_exp<<<edgeB, 256, 0, stream>>>(eexp, erow, E, Etot, rmx, rsum);
    k_spmm<<<edgeW, 256, 0, stream>>>(eexp, rsum, erow, ecol, E, Etot, Wh, hacc);
    k_elu_stats<<<eluB, 128, 0, stream>>>(hacc, N, bnsum, bnsq, 1);
    k_bn_final<<<1, 128, 0, stream>>>(bnsum, bnsq, gamma, beta, bnsc, bnsh, N);

    // ===== layer 1 (BN+ReLU fused into GEMM A-staging) =====
    k_fill<<<1024, 256, 0, stream>>>(out, 0.f, (long long)N * FEAT);
    k_fill<<<256, 256, 0, stream>>>((float*)rmx, 0.f, N);
    k_fill<<<256, 256, 0, stream>>>(rsum, 0.f, N);

    k_gemm<<<mblocks, 256, 0, stream>>>(hacc, W1, Wh, bnsc, bnsh, N);
    k_scores<<<nodeW, 256, 0, stream>>>(Wh, a1, s1, s2, N);
    k_edge_max<<<edgeB, 256, 0, stream>>>(s1, s2, erow, ecol, E, Etot, eexp, rmx);
    k_edge_exp<<<edgeB, 256, 0, stream>>>(eexp, erow, E, Etot, rmx, rsum);
    k_spmm<<<edgeW, 256, 0, stream>>>(eexp, rsum, erow, ecol, E, Etot, Wh, out);
    k_elu_stats<<<eluB, 128, 0, stream>>>(out, N, nullptr, nullptr, 0);
}